// MockBackbone_29918742184132
// MI455X (gfx1250) — compile-verified
//
#include <hip/hip_runtime.h>
#include <stdint.h>
#include <stddef.h>

// Problem constants (match reference).
#define B_ 4
#define L_ 4096
#define V_ 8192
#define SIGMA_SCALE_ 0.01f

typedef float v4f __attribute__((ext_vector_type(4)));
typedef uint32_t u32x4 __attribute__((ext_vector_type(4)));
typedef uint32_t u32x8 __attribute__((ext_vector_type(8)));

// ---------------------------------------------------------------------------
// CDNA5 async global->LDS helpers (inline asm; portable across toolchains).
// One b128 op moves 16 B per lane; LDS destination byte-offset rides in a
// VGPR (low 32 bits of the flat shared-memory address per aperture mapping).
// ---------------------------------------------------------------------------
__device__ __forceinline__ void async_load_b128(const void* gaddr, void* laddr) {
  asm volatile("global_load_async_to_lds_b128 %0, %1, off"
               :
               : "v"((uint32_t)(uintptr_t)laddr), "v"((uint64_t)(uintptr_t)gaddr)
               : "memory");
}
__device__ __forceinline__ void wait_asynccnt_8() {
  asm volatile("s_wait_asynccnt 0x8" ::: "memory");
}
__device__ __forceinline__ void wait_asynccnt_0() {
  asm volatile("s_wait_asynccnt 0x0" ::: "memory");
}

// Issue one 32 KB row fetch: 8 async b128 ops per thread
// (256 threads x 8 x 16 B = 32 KB).
__device__ __forceinline__ void issue_row_async(const float* __restrict__ grow,
                                                float* lbuf, int tx) {
#pragma unroll
  for (int k = 0; k < 8; ++k) {
    const int e4 = tx + k * 256;  // float4 index within the row
    async_load_b128(grow + (size_t)e4 * 4, lbuf + (size_t)e4 * 4);
  }
}

// ---------------------------------------------------------------------------
// Kernel 1: col_sum[o] = sum_i W[o,i] (clause of 8 b128 loads in flight),
// fused into base[b][o] = 0.01*sigma[b]*cs + bias[o]. Deterministic order.
// ---------------------------------------------------------------------------
__global__ __launch_bounds__(256) void colsum_base_kernel(
    const float* __restrict__ W, const float* __restrict__ sigma,
    const float* __restrict__ bias, float* __restrict__ base /* [B_][V_] */) {
  __shared__ float red[256];
  const int o = blockIdx.x;
  const int tx = threadIdx.x;
  const v4f* row4 = (const v4f*)(W + (size_t)o * V_);
  v4f acc = {0.0f, 0.0f, 0.0f, 0.0f};
#pragma unroll
  for (int k = 0; k < 8; ++k) acc += row4[tx + k * 256];
  red[tx] = (acc.x + acc.y) + (acc.z + acc.w);
  __syncthreads();
  for (int w = 128; w >= 1; w >>= 1) {
    if (tx < w) red[tx] += red[tx + w];
    __syncthreads();
  }
  if (tx < B_) {
    base[(size_t)tx * V_ + o] = SIGMA_SCALE_ * sigma[tx] * red[0] + bias[o];
  }
}

// ---------------------------------------------------------------------------
// Kernel 2: transpose WT[v][o] = W[o][v] using the Tensor Data Mover.
// Wave 0 issues one tensor_load_to_lds per 32x32 f32 tile with HW padding
// (1 DWORD every 32 DWORDs -> LDS row stride 33 -> bank-conflict-free
// transposed reads), then all 8 waves write the transposed tile coalesced.
// Descriptor per CDNA5 ISA; groups 2/3 are NULL
// (<=2-D tensor, 2-group short form). Tracked on TENSORcnt.
// ---------------------------------------------------------------------------
__global__ __launch_bounds__(256) void transpose_kernel(
    const float* __restrict__ W, float* __restrict__ WT) {
  __shared__ float tile[32][33];
  const int vo = blockIdx.x * 32;  // column block of W (rows of WT)
  const int oo = blockIdx.y * 32;  // row block of W
  const int tx = threadIdx.x;      // 0..31
  const int ty = threadIdx.y;      // 0..7

  if (ty == 0) {  // wave 0 (wave32, blockDim=(32,8)) issues the TDM fetch
    const uint64_t ga  = (uint64_t)(uintptr_t)(W + (size_t)oo * V_ + vo);
    const uint32_t lds = (uint32_t)(uintptr_t)&tile[0][0];
    u32x4 g0;
    g0[0] = 1u;                                  // count=1, user, no gather
    g0[1] = lds;                                 // lds_addr [63:32]
    g0[2] = (uint32_t)ga;                        // global_addr[31:0]
    g0[3] = (uint32_t)((ga >> 32) & 0x01FFFFFFu) // global_addr[56:32]
          | (2u << 30);                          // type = 2 ("image")
    u32x8 g1;
    g1[0] = (2u << 16)      // data_size = 4 bytes
          | (1u << 20)      // pad_enable
          | (4u << 22)      // pad_interval: 32 DWORDs
          | (0u << 25);     // pad_amount: 1 DWORD
    g1[1] = 32u << 16;      // tensor_dim0 = 32   (bits 79:48 of group)
    g1[2] = 32u << 16;      // tensor_dim0 hi=0 | tensor_dim1 = 32
    g1[3] = 32u << 16;      // tensor_dim1 hi=0 | tile_dim0 = 32
    g1[4] = 32u;            // tile_dim1 = 32 | tile_dim2 = 0
    g1[5] = (uint32_t)V_;   // tensor_dim0_stride = 8192 (row pitch)
    g1[6] = 0u;             // stride hi / tensor_dim1_stride lo
    g1[7] = 0u;             // tensor_dim1_stride hi
    asm volatile("tensor_load_to_lds %0, %1" :: "s"(g0), "s"(g1) : "memory");
    __builtin_amdgcn_s_wait_tensorcnt(0);
  }
  __syncthreads();  // tile resident for all waves

#pragma unroll
  for (int j = 0; j < 4; ++j) {
    const int r = ty + j * 8;
    WT[(size_t)(vo + r) * V_ + (oo + tx)] = tile[tx][r];
  }
}

// ---------------------------------------------------------------------------
// Kernel 3: main gather, double-buffered async global->LDS pipeline.
// 4 tokens per block (same batch b: L_ % 4 == 0). While token i's 32 KB row
// is consumed from LDS (add register-cached base row, non-temporal b128
// store), token i+1's row streams into the other LDS buffer on ASYNCcnt.
// ---------------------------------------------------------------------------
__global__ __launch_bounds__(256) void gather_kernel(
    const int* __restrict__ indices, const float* __restrict__ WT,
    const float* __restrict__ base, float* __restrict__ out) {
  __shared__ float buf[2][V_];  // 2 x 32 KB double buffer
  const int t0 = blockIdx.x * 4;
  const int b  = t0 / L_;
  const int tx = threadIdx.x;

  // Cache this batch's base row in registers (8 x float4 = 32 VGPRs).
  const v4f* base4 = (const v4f*)(base + (size_t)b * V_);
  v4f breg[8];
#pragma unroll
  for (int k = 0; k < 8; ++k) breg[k] = base4[tx + k * 256];

  int idx[4];
#pragma unroll
  for (int i = 0; i < 4; ++i) {
    int v = indices[t0 + i];
    idx[i] = (v < 0) ? 0 : ((v >= V_) ? (V_ - 1) : v);
  }

  // Prologue: start token 0's row into buffer 0.
  issue_row_async(WT + (size_t)idx[0] * V_, buf[0], tx);

#pragma unroll
  for (int i = 0; i < 4; ++i) {
    if (i < 3) {
      // Kick off next row first, then retire current buffer's group of 8
      // (async loads complete in order).
      issue_row_async(WT + (size_t)idx[i + 1] * V_, buf[(i + 1) & 1], tx);
      wait_asynccnt_8();
    } else {
      wait_asynccnt_0();
    }
    __syncthreads();  // all waves' portions of buf[i&1] are resident

    const v4f* l4 = (const v4f*)buf[i & 1];
    v4f* dst4 = (v4f*)(out + (size_t)(t0 + i) * V_);
#pragma unroll
    for (int k = 0; k < 8; ++k) {
      v4f v = l4[tx + k * 256];
      v += breg[k];
      __builtin_nontemporal_store(v, &dst4[tx + k * 256]);
    }
    __syncthreads();  // buffer may now be overwritten by the next issue
  }
}

// ---------------------------------------------------------------------------
// Fallback (only if d_ws can't hold the 256 MB transpose): direct strided
// column gather from W.
// ---------------------------------------------------------------------------
__global__ __launch_bounds__(256) void gather_direct_kernel(
    const int* __restrict__ indices, const float* __restrict__ W,
    const float* __restrict__ base, float* __restrict__ out) {
  const int t = blockIdx.x;
  const int b = t / L_;
  int v = indices[t];
  const int idx = (v < 0) ? 0 : ((v >= V_) ? (V_ - 1) : v);
  const float* baseRow = base + (size_t)b * V_;
  float* dst = out + (size_t)t * V_;
  for (int k = 0; k < V_ / 256; ++k) {
    const int o = threadIdx.x + k * 256;
    float val = W[(size_t)o * V_ + idx] + baseRow[o];
    __builtin_nontemporal_store(val, &dst[o]);
  }
}

// ---------------------------------------------------------------------------
extern "C" void kernel_launch(void* const* d_in, const int* in_sizes, int n_in,
                              void* d_out, int out_size, void* d_ws,
                              size_t ws_size, hipStream_t stream) {
  (void)in_sizes; (void)n_in; (void)out_size;
  const int*   indices = (const int*)d_in[0];    // [B_, L_] int32
  const float* sigma   = (const float*)d_in[1];  // [B_]
  const float* W       = (const float*)d_in[2];  // [V_, V_]
  const float* bias    = (const float*)d_in[3];  // [V_]
  float* out = (float*)d_out;                    // [B_, L_, V_]

  // Workspace layout: base[B_][V_] (128 KB), then WT[V_][V_] (256 MB).
  float* base = (float*)d_ws;
  const size_t base_bytes = (size_t)B_ * V_ * sizeof(float);
  const size_t wt_off = (base_bytes + 255) & ~(size_t)255;
  float* WT = (float*)((char*)d_ws + wt_off);
  const size_t need = wt_off + (size_t)V_ * V_ * sizeof(float);

  colsum_base_kernel<<<V_, 256, 0, stream>>>(W, sigma, bias, base);

  if (ws_size >= need) {
    dim3 tgrid(V_ / 32, V_ / 32);
    transpose_kernel<<<tgrid, dim3(32, 8), 0, stream>>>(W, WT);
    gather_kernel<<<(B_ * L_) / 4, 256, 0, stream>>>(indices, WT, base, out);
  } else {
    gather_direct_kernel<<<B_ * L_, 256, 0, stream>>>(indices, W, base, out);
  }
}